// DualPathLoss_9216999817599
// MI455X (gfx1250) — compile-verified
//
#include <hip/hip_runtime.h>
#include <math.h>

// ---------------------------------------------------------------------------
// DualPathLoss for MI455X (gfx1250, wave32).
// Pipeline (all on `stream`, graph-capture safe, deterministic — no atomics):
//   1. det_cost_kernel      : [8,300,50] cost matrix
//   2. lane_cost_kernel     : [8,32,16] pairwise lane cost via WMMA f32 16x16x4
//   3. hungarian_kernel x2  : rectangular Jonker-Volgenant, 1 block per batch
//   4. det_matched / cls / lane_matched / exist kernels -> per-item partials
//   5. final_kernel         : single-thread deterministic combine -> scalar
// ---------------------------------------------------------------------------

#define B_    8
#define QD_   300
#define GD_   50
#define QL_   32
#define GL_   16
#define NP_   72
#define NSEG_ 71

typedef float v2f __attribute__((ext_vector_type(2)));
typedef float v8f __attribute__((ext_vector_type(8)));

__device__ __forceinline__ float wave_sum32(float v) {
#pragma unroll
  for (int m = 16; m >= 1; m >>= 1) v += __shfl_xor(v, m, 32);
  return v;
}

// =====================  1. detection cost matrix  ==========================
__global__ __launch_bounds__(256) void det_cost_kernel(
    const float* __restrict__ logits,   // [8,300,11]
    const float* __restrict__ boxes,    // [8,300,4] cxcywh
    const int*   __restrict__ gcls,     // [8,50]
    const float* __restrict__ gboxes,   // [8,50,4]
    float* __restrict__ cost)           // [8,300,50]
{
  const int idx = blockIdx.x * blockDim.x + threadIdx.x;  // b*300+q
  if (idx >= B_ * QD_) return;
  const int b = idx / QD_;

  const float* L = logits + (size_t)idx * 11;
  float mx = L[0];
#pragma unroll
  for (int c = 1; c < 11; ++c) mx = fmaxf(mx, L[c]);
  float e[11], s = 0.f;
#pragma unroll
  for (int c = 0; c < 11; ++c) { e[c] = expf(L[c] - mx); s += e[c]; }
  const float inv = 1.f / s;

  const float* pb = boxes + (size_t)idx * 4;
  const float cx = pb[0], cy = pb[1], w = pb[2], h = pb[3];
  const float px1 = cx - 0.5f * w, py1 = cy - 0.5f * h;
  const float px2 = cx + 0.5f * w, py2 = cy + 0.5f * h;
  const float pa = fmaxf(px2 - px1, 0.f) * fmaxf(py2 - py1, 0.f);

  for (int g = 0; g < GD_; ++g) {
    const int   cls   = gcls[b * GD_ + g];
    const float ccost = -e[cls] * inv;
    const float* gb = gboxes + (size_t)(b * GD_ + g) * 4;
    const float l1 = fabsf(cx - gb[0]) + fabsf(cy - gb[1]) +
                     fabsf(w - gb[2]) + fabsf(h - gb[3]);
    const float gx1 = gb[0] - 0.5f * gb[2], gy1 = gb[1] - 0.5f * gb[3];
    const float gx2 = gb[0] + 0.5f * gb[2], gy2 = gb[1] + 0.5f * gb[3];
    const float ga = fmaxf(gx2 - gx1, 0.f) * fmaxf(gy2 - gy1, 0.f);
    const float ix = fmaxf(fminf(px2, gx2) - fmaxf(px1, gx1), 0.f);
    const float iy = fmaxf(fminf(py2, gy2) - fmaxf(py1, gy1), 0.f);
    const float inter = ix * iy;
    const float uni   = pa + ga - inter;
    const float iou   = inter / fmaxf(uni, 1e-6f);
    const float ex = fmaxf(fmaxf(px2, gx2) - fminf(px1, gx1), 0.f);
    const float ey = fmaxf(fmaxf(py2, gy2) - fminf(py1, gy1), 0.f);
    const float enc  = ex * ey;
    const float giou = iou - (enc - uni) / fmaxf(enc, 1e-6f);
    cost[(size_t)idx * GD_ + g] = 2.f * ccost + 5.f * l1 - 2.f * giou;
  }
}

// =====================  2. lane pairwise cost (WMMA)  ======================
// One wave per (b,q,g) pair. For each direction the 72x71 point-to-segment
// distance matrix is produced from two f32 16x16x4 WMMAs per tile:
//   n[i][j] = ap.ab   via A=[px,py,1,0],         B=[abx,aby,-(A.ab),0]
//   q[i][j] = |p-A|^2 via A=[-2px,-2py,|p|^2,1], B=[Ax,Ay,1,|A|^2]
// then d^2 = q - t*(2n - t*den), t = clamp(n*inv_den,0,1).
// inv_den is precomputed at LDS-fill time so the hot loop has no divides,
// and all lane-half operand selection is branchless (v_cndmask) so EXEC
// stays all-ones across the WMMAs.
__global__ __launch_bounds__(256) void lane_cost_kernel(
    const float* __restrict__ pred_pts,      // [8,32,72,2]
    const float* __restrict__ gt_pts,        // [8,16,72,2]
    const float* __restrict__ vis,           // [8,16,72]
    const float* __restrict__ exist_logits,  // [8,32,1]
    float* __restrict__ cost_lane)           // [8,32,16]
{
  __shared__ float smem[8 * 2 * NSEG_ * 8];  // 8 waves * 2 dirs * 71 segs * 8f
  const int lane = threadIdx.x & 31;
  const int wid  = threadIdx.x >> 5;
  const int pair = blockIdx.x * 8 + wid;  // 0..4095 (exactly 512 blocks)
  const int g = pair & 15;
  const int q = (pair >> 4) & 31;
  const int b = pair >> 9;

  const float* P  = pred_pts + ((size_t)(b * QL_ + q)) * NP_ * 2;
  const float* Gp = gt_pts   + ((size_t)(b * GL_ + g)) * NP_ * 2;
  const float* V  = vis      + ((size_t)(b * GL_ + g)) * NP_;

  float* sw = smem + wid * (2 * NSEG_ * 8);
  // Fill segment tables: dir 0 -> segments from GT, dir 1 -> from pred.
  for (int d = 0; d < 2; ++d) {
    const float* S = (d == 0) ? Gp : P;
    float* sd = sw + d * (NSEG_ * 8);
    for (int j = lane; j < NSEG_; j += 32) {
      const float ax = S[2 * j], ay = S[2 * j + 1];
      const float bx = S[2 * j + 2], by = S[2 * j + 3];
      const float abx = bx - ax, aby = by - ay;
      const float den = fmaxf(abx * abx + aby * aby, 1e-8f);
      float* e = sd + j * 8;
      e[0] = ax; e[1] = ay; e[2] = abx; e[3] = aby;
      e[4] = -(ax * abx + ay * aby);
      e[5] = ax * ax + ay * ay;
      e[6] = den;
      e[7] = 1.f / den;          // hot loop: divide -> multiply
    }
  }
  __syncthreads();

  const bool lo = (lane < 16);
  float res[2];
  for (int d = 0; d < 2; ++d) {
    const float* RP = (d == 0) ? P : Gp;  // row (query) points
    const float* sd = sw + d * (NSEG_ * 8);
    float sum_d = 0.f, sum_w = 0.f;
    for (int ti = 0; ti < 5; ++ti) {
      const int  m   = ti * 16 + (lane & 15);
      const bool mok = (m < NP_);
      const int  mc  = mok ? m : (NP_ - 1);
      float px = RP[2 * mc], py = RP[2 * mc + 1];
      px = mok ? px : 0.f;
      py = mok ? py : 0.f;
      const float p2 = px * px + py * py;
      // A operands: lanes 0-15 carry K0,K1 of row m; lanes 16-31 carry K2,K3.
      v2f aN, aQ;
      aN[0] = lo ? px : 1.f;
      aN[1] = lo ? py : 0.f;
      aQ[0] = lo ? (-2.f * px) : p2;
      aQ[1] = lo ? (-2.f * py) : 1.f;

      float rowmin[8];
#pragma unroll
      for (int r = 0; r < 8; ++r) rowmin[r] = 1e30f;

      for (int tj = 0; tj < 5; ++tj) {
        const int  j   = tj * 16 + (lane & 15);
        const bool jok = (j < NSEG_);
        const int  jc  = jok ? j : (NSEG_ - 1);
        const float* e = sd + jc * 8;
        const float ax = e[0], ay = e[1], abx = e[2], aby = e[3];
        const float naab = e[4], a2 = e[5], den = e[6], iden = e[7];
        // B operands: VGPR0 lane-halves hold rows {K0,K2}, VGPR1 {K1,K3}.
        v2f bN, bQ;
        bN[0] = lo ? abx : naab;
        bN[1] = lo ? aby : 0.f;
        bQ[0] = lo ? ax : 1.f;
        bQ[1] = lo ? ay : a2;
        v8f z = {};
        v8f nmat = __builtin_amdgcn_wmma_f32_16x16x4_f32(
            false, aN, false, bN, (short)0, z, false, false);
        v8f qmat = __builtin_amdgcn_wmma_f32_16x16x4_f32(
            false, aQ, false, bQ, (short)0, z, false, false);
#pragma unroll
        for (int r = 0; r < 8; ++r) {
          const float nn = nmat[r], qq = qmat[r];
          float t  = fminf(fmaxf(nn * iden, 0.f), 1.f);
          float d2 = qq - t * (2.f * nn - t * den);
          d2 = fmaxf(d2, 0.f);
          d2 = jok ? d2 : 1e30f;
          rowmin[r] = fminf(rowmin[r], d2);
        }
      }
      // Min over the 16 columns of each half-wave (rows r / r+8).
#pragma unroll
      for (int mask = 1; mask <= 8; mask <<= 1)
#pragma unroll
        for (int r = 0; r < 8; ++r)
          rowmin[r] = fminf(rowmin[r], __shfl_xor(rowmin[r], mask, 32));
      if ((lane & 15) == 0) {
        const int mbase = ti * 16 + ((lane >= 16) ? 8 : 0);
#pragma unroll
        for (int r = 0; r < 8; ++r) {
          const int mm = mbase + r;
          if (mm < NP_) {
            const float wgt = (d == 0) ? 1.f : V[mm];
            sum_d += wgt * sqrtf(rowmin[r]);
            sum_w += wgt;
          }
        }
      }
    }
    sum_d = wave_sum32(sum_d);
    sum_w = wave_sum32(sum_w);
    res[d] = sum_d / fmaxf(sum_w, 1.f);  // d0: mean over 72; d1: vis-weighted
  }

  // Tangent term (once per pair).
  float accT = 0.f, accS = 0.f;
  for (int j = lane; j < NSEG_; j += 32) {
    const float ptx = P[2 * j + 2] - P[2 * j],  pty = P[2 * j + 3] - P[2 * j + 1];
    const float gtx = Gp[2 * j + 2] - Gp[2 * j], gty = Gp[2 * j + 3] - Gp[2 * j + 1];
    const float pl = fmaxf(sqrtf(ptx * ptx + pty * pty), 1e-6f);
    const float gl = fmaxf(sqrtf(gtx * gtx + gty * gty), 1e-6f);
    const float dt = (ptx * gtx + pty * gty) / (pl * gl);
    const float sv = V[j] * V[j + 1];
    accT += (1.f - dt) * sv;
    accS += sv;
  }
  accT = wave_sum32(accT);
  accS = wave_sum32(accS);
  if (lane == 0) {
    const float x   = exist_logits[b * QL_ + q];
    const float sig = 1.f / (1.f + expf(-x));
    cost_lane[pair] = 0.5f * (res[0] + res[1]) + 0.35f * (accT / fmaxf(accS, 1.f))
                      - 0.25f * sig;
  }
}

// =====================  3. rectangular Hungarian (JV)  =====================
// Solves min-cost assignment of G columns of cost[Q][G] to distinct rows;
// equivalent to the reference's zero-padded 300x300 square solve restricted
// to the first G columns. One block per batch; lanes parallelize the O(Q)
// slack scan + argmin; serial path/potential logic mirrors the reference
// (including first-min-index tie-breaking).
#define HBS 128
__global__ __launch_bounds__(HBS) void hungarian_kernel(
    const float* __restrict__ cost, int* __restrict__ pi, int Q, int G)
{
  const int b = blockIdx.x;
  const float* a = cost + (size_t)b * Q * G;  // a(q,g) = a[q*G+g]
  int* piB = pi + b * G;
  const int t = threadIdx.x;

  __shared__ float u[64];                 // rows: G+1 <= 51
  __shared__ float v[QD_ + 4], minv[QD_ + 4];
  __shared__ int   p[QD_ + 4], way[QD_ + 4];
  __shared__ unsigned char used[QD_ + 4];
  __shared__ float redv[HBS];
  __shared__ int   redj[HBS];
  __shared__ int   s_j0;

  for (int j = t; j <= Q; j += HBS) { v[j] = 0.f; p[j] = 0; }
  for (int i = t; i <= G; i += HBS) u[i] = 0.f;
  __syncthreads();

  for (int i = 1; i <= G; ++i) {
    if (t == 0) { p[0] = i; s_j0 = 0; }
    for (int j = t; j <= Q; j += HBS) { minv[j] = 1e30f; used[j] = 0; }
    __syncthreads();
    while (true) {
      if (t == 0) used[s_j0] = 1;
      __syncthreads();
      const int j0 = s_j0;
      const int i0 = p[j0];
      const float ui0 = u[i0];
      float bv = 1e30f; int bj = Q + 1;
      for (int j = 1 + t; j <= Q; j += HBS) {
        if (!used[j]) {
          const float cur = a[(size_t)(j - 1) * G + (i0 - 1)] - ui0 - v[j];
          float mv = minv[j];
          if (cur < mv) { mv = cur; minv[j] = cur; way[j] = j0; }
          if (mv < bv || (mv == bv && j < bj)) { bv = mv; bj = j; }
        }
      }
      redv[t] = bv; redj[t] = bj;
      __syncthreads();
      for (int s = HBS / 2; s > 0; s >>= 1) {
        if (t < s) {
          const float ov = redv[t + s]; const int oj = redj[t + s];
          if (ov < redv[t] || (ov == redv[t] && oj < redj[t])) {
            redv[t] = ov; redj[t] = oj;
          }
        }
        __syncthreads();
      }
      const float delta = redv[0];
      const int   j1    = redj[0];
      __syncthreads();
      for (int j = t; j <= Q; j += HBS) {
        if (used[j]) { u[p[j]] += delta; v[j] -= delta; }  // distinct p[j]
        else         { minv[j] -= delta; }
      }
      if (t == 0) s_j0 = j1;
      __syncthreads();
      if (p[s_j0] == 0) break;
    }
    if (t == 0) {  // augment
      int j0 = s_j0;
      while (j0 != 0) { const int j1 = way[j0]; p[j0] = p[j1]; j0 = j1; }
    }
    __syncthreads();
  }
  for (int j = 1 + t; j <= Q; j += HBS)
    if (p[j] != 0) piB[p[j] - 1] = j - 1;
}

// =====================  4. matched losses (partials)  ======================
__global__ __launch_bounds__(256) void det_matched_kernel(
    const float* __restrict__ boxes, const float* __restrict__ gboxes,
    const int* __restrict__ pi_det, float* __restrict__ part /*[800]*/)
{
  const int idx = blockIdx.x * blockDim.x + threadIdx.x;  // b*50+g
  if (idx >= B_ * GD_) return;
  const int b = idx / GD_;
  const int q = pi_det[idx];
  const float* pb = boxes + (size_t)(b * QD_ + q) * 4;
  const float* gb = gboxes + (size_t)idx * 4;
  const float l1 = fabsf(pb[0] - gb[0]) + fabsf(pb[1] - gb[1]) +
                   fabsf(pb[2] - gb[2]) + fabsf(pb[3] - gb[3]);
  const float px1 = pb[0] - 0.5f * pb[2], py1 = pb[1] - 0.5f * pb[3];
  const float px2 = pb[0] + 0.5f * pb[2], py2 = pb[1] + 0.5f * pb[3];
  const float gx1 = gb[0] - 0.5f * gb[2], gy1 = gb[1] - 0.5f * gb[3];
  const float gx2 = gb[0] + 0.5f * gb[2], gy2 = gb[1] + 0.5f * gb[3];
  const float pa = fmaxf(px2 - px1, 0.f) * fmaxf(py2 - py1, 0.f);
  const float ga = fmaxf(gx2 - gx1, 0.f) * fmaxf(gy2 - gy1, 0.f);
  const float ix = fmaxf(fminf(px2, gx2) - fmaxf(px1, gx1), 0.f);
  const float iy = fmaxf(fminf(py2, gy2) - fmaxf(py1, gy1), 0.f);
  const float inter = ix * iy, uni = pa + ga - inter;
  const float iou = inter / fmaxf(uni, 1e-6f);
  const float ex = fmaxf(fmaxf(px2, gx2) - fminf(px1, gx1), 0.f);
  const float ey = fmaxf(fmaxf(py2, gy2) - fminf(py1, gy1), 0.f);
  const float enc = ex * ey;
  const float giou = iou - (enc - uni) / fmaxf(enc, 1e-6f);
  part[idx] = l1;
  part[B_ * GD_ + idx] = 1.f - giou;
}

__global__ __launch_bounds__(256) void cls_kernel(
    const float* __restrict__ logits, const int* __restrict__ gcls,
    const int* __restrict__ pi_det, float* __restrict__ part /*[4800]*/)
{
  const int idx = blockIdx.x * blockDim.x + threadIdx.x;  // b*300+q
  if (idx >= B_ * QD_) return;
  const int b = idx / QD_, q = idx % QD_;
  int tgt = 10;
  for (int g = 0; g < GD_; ++g)
    if (pi_det[b * GD_ + g] == q) tgt = gcls[b * GD_ + g];
  const float* L = logits + (size_t)idx * 11;
  float mx = L[0];
#pragma unroll
  for (int c = 1; c < 11; ++c) mx = fmaxf(mx, L[c]);
  float s = 0.f;
#pragma unroll
  for (int c = 0; c < 11; ++c) s += expf(L[c] - mx);
  const float nll = -(L[tgt] - mx - logf(s));
  const float w = (tgt == 10) ? 0.1f : 1.f;
  part[idx] = w * nll;
  part[B_ * QD_ + idx] = w;
}

__device__ __forceinline__ float p2seg(float px, float py, float ax, float ay,
                                       float bx, float by) {
  const float abx = bx - ax, aby = by - ay;
  const float den = fmaxf(abx * abx + aby * aby, 1e-8f);
  const float t = fminf(fmaxf(((px - ax) * abx + (py - ay) * aby) / den, 0.f), 1.f);
  const float dx = px - (ax + t * abx), dy = py - (ay + t * aby);
  return sqrtf(dx * dx + dy * dy);
}

__global__ __launch_bounds__(32) void lane_matched_kernel(
    const float* __restrict__ pred_pts, const float* __restrict__ gt_pts,
    const float* __restrict__ vis, const float* __restrict__ type_logits,
    const int* __restrict__ gt_type, const int* __restrict__ pi_lane,
    float* __restrict__ part /*[128*5]*/)
{
  const int bg = blockIdx.x;              // b*16+g
  const int b = bg >> 4;
  const int lane = threadIdx.x;
  const int q = pi_lane[bg];
  const float* P  = pred_pts + ((size_t)(b * QL_ + q)) * NP_ * 2;
  const float* Gp = gt_pts + (size_t)bg * NP_ * 2;
  const float* V  = vis + (size_t)bg * NP_;

  float s1 = 0.f, sv = 0.f, sdpg = 0.f, sdgp = 0.f, svg = 0.f;
  for (int m = lane; m < NP_; m += 32) {
    const float d = fabsf(P[2 * m] - Gp[2 * m]) + fabsf(P[2 * m + 1] - Gp[2 * m + 1]);
    s1 += d * V[m];
    sv += V[m];
    float best = 1e30f;
    for (int j = 0; j < NSEG_; ++j)
      best = fminf(best, p2seg(P[2 * m], P[2 * m + 1], Gp[2 * j], Gp[2 * j + 1],
                               Gp[2 * j + 2], Gp[2 * j + 3]));
    sdpg += best;
    best = 1e30f;
    for (int j = 0; j < NSEG_; ++j)
      best = fminf(best, p2seg(Gp[2 * m], Gp[2 * m + 1], P[2 * j], P[2 * j + 1],
                               P[2 * j + 2], P[2 * j + 3]));
    sdgp += best * V[m];
    svg += V[m];
  }
  float at = 0.f, as = 0.f, am = 0.f, ac = 0.f;
  for (int j = lane; j < NSEG_; j += 32) {
    const float ptx = P[2 * j + 2] - P[2 * j],  pty = P[2 * j + 3] - P[2 * j + 1];
    const float gtx = Gp[2 * j + 2] - Gp[2 * j], gty = Gp[2 * j + 3] - Gp[2 * j + 1];
    const float pl = fmaxf(sqrtf(ptx * ptx + pty * pty), 1e-6f);
    const float gl = fmaxf(sqrtf(gtx * gtx + gty * gty), 1e-6f);
    const float svj = V[j] * V[j + 1];
    at += (1.f - (ptx * gtx + pty * gty) / (pl * gl)) * svj;
    as += svj;
    if (j < NP_ - 2) {
      const float sx = P[2 * j + 4] - 2.f * P[2 * j + 2] + P[2 * j];
      const float sy = P[2 * j + 5] - 2.f * P[2 * j + 3] + P[2 * j + 1];
      const float cv = V[j] * V[j + 1] * V[j + 2];
      am += sqrtf(sx * sx + sy * sy) * cv;
      ac += cv;
    }
  }
  s1 = wave_sum32(s1);   sv = wave_sum32(sv);
  sdpg = wave_sum32(sdpg); sdgp = wave_sum32(sdgp); svg = wave_sum32(svg);
  at = wave_sum32(at);   as = wave_sum32(as);
  am = wave_sum32(am);   ac = wave_sum32(ac);
  if (lane == 0) {
    const float* tl = type_logits + (size_t)(b * QL_ + q) * 7;
    float mx = tl[0];
#pragma unroll
    for (int c = 1; c < 7; ++c) mx = fmaxf(mx, tl[c]);
    float se = 0.f;
#pragma unroll
    for (int c = 0; c < 7; ++c) se += expf(tl[c] - mx);
    float* o = part + bg * 5;
    o[0] = s1 / fmaxf(sv, 1.f);
    o[1] = 0.5f * (sdpg / (float)NP_ + sdgp / fmaxf(svg, 1.f));
    o[2] = at / fmaxf(as, 1.f);
    o[3] = am / fmaxf(ac, 1.f);
    o[4] = -(tl[gt_type[bg]] - mx - logf(se));
  }
}

__global__ __launch_bounds__(256) void exist_kernel(
    const float* __restrict__ exist_logits, const int* __restrict__ pi_lane,
    float* __restrict__ part /*[256]*/)
{
  const int idx = threadIdx.x;  // b*32+q
  if (idx >= B_ * QL_) return;
  const int b = idx / QL_, q = idx % QL_;
  float tgt = 0.f;
  for (int g = 0; g < GL_; ++g)
    if (pi_lane[b * GL_ + g] == q) tgt = 1.f;
  const float x = exist_logits[idx];
  const float sp = (x > 0.f) ? (x + log1pf(expf(-x))) : log1pf(expf(x));
  part[idx] = sp - x * tgt;
}

// =====================  5. deterministic final combine  ====================
__global__ void final_kernel(const float* __restrict__ part_det,
                             const float* __restrict__ part_cls,
                             const float* __restrict__ part_lane,
                             const float* __restrict__ part_exist,
                             const float* __restrict__ has_lanes,
                             float* __restrict__ out)
{
  float sl1 = 0.f, sgiou = 0.f;
  for (int i = 0; i < B_ * GD_; ++i) { sl1 += part_det[i]; sgiou += part_det[B_ * GD_ + i]; }
  float swn = 0.f, sw = 0.f;
  for (int i = 0; i < B_ * QD_; ++i) { swn += part_cls[i]; sw += part_cls[B_ * QD_ + i]; }
  float lp = 0.f, lc = 0.f, ld = 0.f, ls = 0.f, lt = 0.f;
  for (int i = 0; i < B_ * GL_; ++i) {
    lp += part_lane[i * 5 + 0]; lc += part_lane[i * 5 + 1];
    ld += part_lane[i * 5 + 2]; ls += part_lane[i * 5 + 3];
    lt += part_lane[i * 5 + 4];
  }
  float se = 0.f;
  for (int i = 0; i < B_ * QL_; ++i) se += part_exist[i];

  const float n_det = (float)(B_ * GD_), n_lane = (float)(B_ * GL_);
  const float det_total = 2.f * (swn / sw) + 5.f * (sl1 / n_det) + 2.f * (sgiou / n_det);
  const float lane_total = 1.5f * (se / (float)(B_ * QL_)) + 3.f * (lp / n_lane) +
                           0.5f * (lt / n_lane) + 5.f * (lc / n_lane) +
                           1.5f * (ld / n_lane) + 0.25f * (ls / n_lane);
  float gate = 0.f;
  for (int b = 0; b < B_; ++b)
    if (has_lanes[b] > 0.5f) gate = 1.f;
  out[0] = det_total + gate * lane_total;
}

// ===========================================================================
extern "C" void kernel_launch(void* const* d_in, const int* in_sizes, int n_in,
                              void* d_out, int out_size, void* d_ws, size_t ws_size,
                              hipStream_t stream) {
  const float* det_logits  = (const float*)d_in[0];
  const float* det_boxes   = (const float*)d_in[1];
  const int*   gt_classes  = (const int*)d_in[2];
  const float* gt_boxes    = (const float*)d_in[3];
  const float* exist_log   = (const float*)d_in[4];
  const float* pred_pts    = (const float*)d_in[5];
  const float* type_logits = (const float*)d_in[6];
  const float* gt_pts      = (const float*)d_in[8];
  const float* gt_vis      = (const float*)d_in[9];
  const int*   gt_type     = (const int*)d_in[10];
  const float* has_lanes   = (const float*)d_in[11];
  float* out = (float*)d_out;

  char* wsb = (char*)d_ws;
  float* cost_det   = (float*)(wsb);                       // 120000 f
  float* cost_lane  = (float*)(wsb + 480000);              // 4096 f
  float* part_det   = (float*)(wsb + 496384);              // 800 f
  float* part_cls   = (float*)(wsb + 499584);              // 4800 f
  float* part_lane  = (float*)(wsb + 518784);              // 640 f
  float* part_exist = (float*)(wsb + 521344);              // 256 f
  int*   pi_det     = (int*)  (wsb + 522368);              // 400 i
  int*   pi_lane    = (int*)  (wsb + 523968);              // 128 i
  (void)in_sizes; (void)n_in; (void)out_size; (void)ws_size;

  det_cost_kernel<<<(B_ * QD_ + 255) / 256, 256, 0, stream>>>(
      det_logits, det_boxes, gt_classes, gt_boxes, cost_det);
  lane_cost_kernel<<<512, 256, 0, stream>>>(
      pred_pts, gt_pts, gt_vis, exist_log, cost_lane);
  hungarian_kernel<<<B_, HBS, 0, stream>>>(cost_det, pi_det, QD_, GD_);
  hungarian_kernel<<<B_, HBS, 0, stream>>>(cost_lane, pi_lane, QL_, GL_);
  det_matched_kernel<<<(B_ * GD_ + 255) / 256, 256, 0, stream>>>(
      det_boxes, gt_boxes, pi_det, part_det);
  cls_kernel<<<(B_ * QD_ + 255) / 256, 256, 0, stream>>>(
      det_logits, gt_classes, pi_det, part_cls);
  lane_matched_kernel<<<B_ * GL_, 32, 0, stream>>>(
      pred_pts, gt_pts, gt_vis, type_logits, gt_type, pi_lane, part_lane);
  exist_kernel<<<1, 256, 0, stream>>>(exist_log, pi_lane, part_exist);
  final_kernel<<<1, 1, 0, stream>>>(part_det, part_cls, part_lane, part_exist,
                                    has_lanes, out);
}